// LSS_Core_5222680232625
// MI455X (gfx1250) — compile-verified
//
#include <hip/hip_runtime.h>
#include <math.h>

// ---- problem constants (from reference) ----
#define BN      24          // B*N
#define CIN     256
#define IMG_H   16
#define IMG_W   44
#define HW      704         // IMG_H*IMG_W
#define DDEPTH  41
#define CC      64
#define MOUT    105         // DDEPTH + CC
#define XMAX    200
#define YMAX    200
#define VGRID   40000       // XMAX*YMAX
#define NPOINTS (BN * DDEPTH * HW)   // 692736

typedef __attribute__((ext_vector_type(2))) float v2f;
typedef __attribute__((ext_vector_type(8))) float v8f;

// -----------------------------------------------------------------------------
// Kernel 1: feat = W(105x256) @ x(256x704) + bias, per image, via f32 WMMA.
// One wave computes one 16x16 tile of feat. K stepped by 4 (V_WMMA_F32_16X16X4_F32),
// 4 K-steps batched per loop iteration so all loads clause together and the
// 4 WMMAs issue back-to-back while the next group's loads are in flight.
//
// A-frag (16x4, MxK): lanes 0-15 hold M=lane, K={0,1}; lanes 16-31 M=lane-16, K={2,3}.
// B-frag (4x16, KxN): lanes 0-15 hold N=lane, K={0,1}; lanes 16-31 N=lane-16, K={2,3}.
// D-frag (16x16 f32): VGPR r, lanes 0-15 -> M=r; lanes 16-31 -> M=r+8; N=lane%16.
//
// Out-of-range A rows (tile rows 105..111 of the last M-tile) are CLAMPED to a
// valid row instead of predicated: they compute garbage D rows that are never
// stored, which keeps the loop branch-free and EXEC all-1s (WMMA requirement).
// -----------------------------------------------------------------------------
__global__ void feat_gemm_wmma(const float* __restrict__ x,     // [BN][CIN][HW]
                               const float* __restrict__ W,     // [MOUT][CIN]
                               const float* __restrict__ bias,  // [MOUT]
                               float* __restrict__ feat)        // [BN][MOUT][HW]
{
    const int MT = 7;   // ceil(105/16)
    const int NT = 44;  // 704/16
    const int lane = threadIdx.x & 31;
    const int wave = threadIdx.x >> 5;
    const int gid  = blockIdx.x * (blockDim.x >> 5) + wave;
    if (gid >= BN * MT * NT) return;          // wave-uniform: EXEC stays all-1s

    const int img = gid / (MT * NT);
    const int rem = gid % (MT * NT);
    const int mt  = rem / NT;
    const int nt  = rem % NT;

    const int half = lane >> 4;               // 0: K-pair {0,1}; 1: K-pair {2,3}
    const int l16  = lane & 15;
    const int m_raw = mt * 16 + l16;          // A-matrix row for this lane
    const int m    = (m_raw < MOUT) ? m_raw : (MOUT - 1);   // clamp (branch-free)
    const int col  = nt * 16 + l16;           // B/D column (pixel)

    const float* Arow = W + (size_t)m * CIN + half * 2;          // 8B aligned
    const float* Bcol = x + (size_t)img * CIN * HW
                          + (size_t)(half * 2) * HW + col;

    v8f acc = {};
    for (int k0 = 0; k0 < CIN; k0 += 16) {
        v2f a[4];
        v2f b[4];
#pragma unroll
        for (int u = 0; u < 4; ++u) {
            const int k = k0 + u * 4;
            a[u] = *(const v2f*)(Arow + k);                      // global_load_b64
            b[u].x = Bcol[(size_t)k * HW];                       // coalesced b32
            b[u].y = Bcol[(size_t)(k + 1) * HW];
        }
#pragma unroll
        for (int u = 0; u < 4; ++u) {
            acc = __builtin_amdgcn_wmma_f32_16x16x4_f32(
                      /*neg_a=*/false, a[u], /*neg_b=*/false, b[u],
                      /*c_mod=*/(short)0, acc,
                      /*reuse_a=*/false, /*reuse_b=*/false);
        }
    }

    float* fimg = feat + (size_t)img * MOUT * HW;
#pragma unroll
    for (int r = 0; r < 8; ++r) {
        const int mo = mt * 16 + half * 8 + r;
        if (mo < MOUT) fimg[(size_t)mo * HW + col] = acc[r] + bias[mo];
    }
}

// -----------------------------------------------------------------------------
// Kernel 2: in-place softmax over the 41 depth channels, one thread per pixel.
// Reads/writes are coalesced across the wave (pix is the fast dimension).
// -----------------------------------------------------------------------------
__global__ void depth_softmax(float* __restrict__ feat)   // [BN][MOUT][HW]
{
    const int t = blockIdx.x * blockDim.x + threadIdx.x;
    if (t >= BN * HW) return;
    const int img = t / HW;
    const int pix = t % HW;

    float* f = feat + (size_t)img * MOUT * HW + pix;

    float mx = -INFINITY;
#pragma unroll
    for (int d = 0; d < DDEPTH; ++d) mx = fmaxf(mx, f[(size_t)d * HW]);

    float s = 0.0f;
#pragma unroll
    for (int d = 0; d < DDEPTH; ++d) {
        const float e = expf(f[(size_t)d * HW] - mx);
        f[(size_t)d * HW] = e;
        s += e;
    }
    const float inv = 1.0f / s;
#pragma unroll
    for (int d = 0; d < DDEPTH; ++d) f[(size_t)d * HW] *= inv;
}

// -----------------------------------------------------------------------------
// Kernel 3: transpose context channels to pixel-major [img][pix][CC] so the
// scatter wave reads 64 contiguous floats per point. Reads coalesced over pix.
// -----------------------------------------------------------------------------
__global__ void ctx_transpose(const float* __restrict__ feat,  // [BN][MOUT][HW]
                              float* __restrict__ ctx)         // [BN][HW][CC]
{
    const int t = blockIdx.x * blockDim.x + threadIdx.x;
    if (t >= BN * CC * HW) return;
    const int pix = t % HW;
    const int r   = t / HW;
    const int c   = r % CC;
    const int img = r / CC;
    ctx[((size_t)img * HW + pix) * CC + c] =
        feat[(size_t)img * MOUT * HW + (size_t)(DDEPTH + c) * HW + pix];
}

// -----------------------------------------------------------------------------
// Kernel 4: voxel splat. One wave per point; each lane handles 2 channels.
// Per point: broadcast depth prob, 64 contiguous context floats, voxel index
// from geom, 64 global f32 atomics (non-returning -> global_atomic_add_f32,
// STOREcnt). BEV accumulator (10.2MB) stays resident in the 192MB L2.
// Early-outs are wave-uniform (whole point in/out of grid).
// -----------------------------------------------------------------------------
__global__ void splat_scatter(const float* __restrict__ geom,  // [NPOINTS][3]
                              const float* __restrict__ feat,  // probs in rows 0..40
                              const float* __restrict__ ctx,   // [BN][HW][CC]
                              float* __restrict__ out)         // [CC][VGRID]
{
    const int lane  = threadIdx.x & 31;
    const int point = blockIdx.x * (blockDim.x >> 5) + (threadIdx.x >> 5);
    if (point >= NPOINTS) return;

    const int img = point / (DDEPTH * HW);
    const int rem = point % (DDEPTH * HW);
    const int d   = rem / HW;
    const int pix = rem % HW;

    const float* g = geom + (size_t)point * 3;
    const float gx = g[0], gy = g[1], gz = g[2];

    // cx = gx*(100-2) - 50 + 1, etc. (matches reference constants)
    const float cx = gx * 98.0f - 49.0f;
    const float cy = gy * 98.0f - 49.0f;
    const float cz = gz * 19.8f - 9.9f;

    const int xi = (int)floorf((cx + 50.0f) * 2.0f);   // /0.5
    const int yi = (int)floorf((cy + 50.0f) * 2.0f);
    const int zi = (int)floorf((cz + 10.0f) * 0.05f);  // /20
    if (xi < 0 || xi >= XMAX || yi < 0 || yi >= YMAX || zi < 0 || zi >= 1) return;

    const float prob = feat[(size_t)img * MOUT * HW + (size_t)d * HW + pix];
    const float* cp  = ctx + ((size_t)img * HW + pix) * CC;

    const int idx = xi * YMAX + yi;
    float* o = out + idx;

    const float v0 = prob * cp[lane];
    const float v1 = prob * cp[lane + 32];
    atomicAdd(o + (size_t)lane        * VGRID, v0);
    atomicAdd(o + (size_t)(lane + 32) * VGRID, v1);
}

// -----------------------------------------------------------------------------
// Launch
// -----------------------------------------------------------------------------
extern "C" void kernel_launch(void* const* d_in, const int* in_sizes, int n_in,
                              void* d_out, int out_size, void* d_ws, size_t ws_size,
                              hipStream_t stream)
{
    const float* x      = (const float*)d_in[0];  // [B,N,CIN,H,W]
    // d_in[1..3] = rots, trans, intrinsics: unused (geometry comes from geom_rand)
    const float* geom   = (const float*)d_in[4];  // [B,N,D,H,W,3]
    const float* conv_w = (const float*)d_in[5];  // [MOUT,CIN]
    const float* conv_b = (const float*)d_in[6];  // [MOUT]
    float* out = (float*)d_out;                   // [1,CC,XMAX,YMAX]

    // workspace layout
    float* feat = (float*)d_ws;                               // BN*MOUT*HW floats (7.1 MB)
    float* ctx  = feat + (size_t)BN * MOUT * HW;              // BN*HW*CC floats  (4.3 MB)

    // zero the BEV accumulator (capture-safe async memset)
    hipMemsetAsync(d_out, 0, (size_t)out_size * sizeof(float), stream);

    // 1) GEMM: 24 images * 7 * 44 tiles = 7392 waves, 8 waves/block
    {
        const int waves  = BN * 7 * 44;
        const int blocks = (waves + 7) / 8;
        feat_gemm_wmma<<<blocks, 256, 0, stream>>>(x, conv_w, conv_b, feat);
    }
    // 2) depth softmax: one thread per pixel
    {
        const int n = BN * HW;
        depth_softmax<<<(n + 255) / 256, 256, 0, stream>>>(feat);
    }
    // 3) context transpose to pixel-major
    {
        const int n = BN * CC * HW;
        ctx_transpose<<<(n + 255) / 256, 256, 0, stream>>>(feat, ctx);
    }
    // 4) scatter: one wave per point, 8 waves/block
    {
        const int blocks = (NPOINTS + 7) / 8;
        splat_scatter<<<blocks, 256, 0, stream>>>(geom, feat, ctx, out);
    }
}